// GAT_15994458210581
// MI455X (gfx1250) — compile-verified
//
#include <hip/hip_runtime.h>
#include <hip/hip_bf16.h>
#include <math.h>

#define N_NODES 100000
#define N_EDGES 1600000
#define ET      (N_EDGES + N_NODES)   // edges + self loops
#define N_GRAPHS 100
#define HEADS 8
#define OUT_CH 16
#define IN_FEAT 128
#define OUTC 128                      // HEADS*OUT_CH
#define EPS 1e-5f
#define NEG_SLOPE 0.2f

typedef __attribute__((ext_vector_type(2))) float v2f;
typedef __attribute__((ext_vector_type(8))) float v8f;

// ---------------- workspace layout (floats) ----------------
// [0,128)        sum_acc
// [128,256)      sq_acc
// [256,384)      scale
// [384,512)      shift
// [512, 512+N*128)                 h
// then N*8 a_src, N*8 a_dst, N*8 mmax, N*8 ssum, N*16 out_nodes,
// 1600 gsum, 100 gcnt
#define WS_SUM   0
#define WS_SQ    128
#define WS_SCALE 256
#define WS_SHIFT 384
#define WS_H     512
#define WS_ASRC  (WS_H + (size_t)N_NODES*128)
#define WS_ADST  (WS_ASRC + (size_t)N_NODES*8)
#define WS_MMAX  (WS_ADST + (size_t)N_NODES*8)
#define WS_SSUM  (WS_MMAX + (size_t)N_NODES*8)
#define WS_ONODE (WS_SSUM + (size_t)N_NODES*8)
#define WS_GSUM  (WS_ONODE + (size_t)N_NODES*16)
#define WS_GCNT  (WS_GSUM + 1600)

__device__ inline void atomicMaxFloat(float* addr, float val) {
    // sign-aware trick; memory initialized to -FLT_MAX
    if (val >= 0.0f) atomicMax((int*)addr, __float_as_int(val));
    else             atomicMin((unsigned int*)addr, (unsigned int)__float_as_int(val));
}

// ---------------- init ----------------
__global__ void __launch_bounds__(256) gat_init(float* ws) {
    int i = blockIdx.x * 256 + threadIdx.x;
    if (i < N_NODES * 16) ws[WS_ONODE + i] = 0.0f;
    if (i < N_NODES * 8) {
        ws[WS_SSUM + i] = 0.0f;
        ws[WS_MMAX + i] = -3.402823466e38f;
    }
    if (i < 1600) ws[WS_GSUM + i] = 0.0f;
    if (i < 100)  ws[WS_GCNT + i] = 0.0f;
    if (i < 128) { ws[WS_SUM + i] = 0.0f; ws[WS_SQ + i] = 0.0f; }
}

// ---------------- BN column statistics ----------------
__global__ void __launch_bounds__(128) gat_bn_stats(const float* __restrict__ x,
                                                    float* __restrict__ ws) {
    const int col = threadIdx.x;               // 0..127
    int r0 = blockIdx.x * 256;
    int r1 = r0 + 256; if (r1 > N_NODES) r1 = N_NODES;
    float s = 0.0f, q = 0.0f;
    for (int r = r0; r < r1; ++r) {
        float v = x[(size_t)r * IN_FEAT + col];
        s += v; q += v * v;
    }
    atomicAdd(&ws[WS_SUM + col], s);
    atomicAdd(&ws[WS_SQ  + col], q);
}

__global__ void __launch_bounds__(128) gat_bn_finalize(const float* __restrict__ gamma,
                                                       const float* __restrict__ beta,
                                                       float* __restrict__ ws) {
    int c = threadIdx.x;
    float mean = ws[WS_SUM + c] * (1.0f / (float)N_NODES);
    float var  = ws[WS_SQ  + c] * (1.0f / (float)N_NODES) - mean * mean;
    float rs   = rsqrtf(var + EPS);
    float sc   = rs * gamma[c];
    ws[WS_SCALE + c] = sc;
    ws[WS_SHIFT + c] = beta[c] - mean * sc;
}

// ---------------- fused BN-apply + GEMM via WMMA fp32 ----------------
// One block = 16 rows x 128 cols of h; 8 waves, each wave one 16x16 tile.
__global__ void __launch_bounds__(256) gat_gemm_wmma(const float* __restrict__ x,
                                                     const float* __restrict__ W,
                                                     const float* __restrict__ ws_ro,
                                                     float* __restrict__ h) {
    __shared__ float ldsA[16][132];            // padded: 132%64=4 -> conflict free
    const float* scale = ws_ro + WS_SCALE;
    const float* shift = ws_ro + WS_SHIFT;
    const int rowBase = blockIdx.x * 16;       // 6250 blocks, exact

    for (int i = threadIdx.x; i < 16 * IN_FEAT; i += 256) {
        int r = i >> 7, c = i & 127;
        float v = x[(size_t)(rowBase + r) * IN_FEAT + c];
        ldsA[r][c] = v * scale[c] + shift[c];
    }
    __syncthreads();

    const int wave = threadIdx.x >> 5;         // 0..7
    const int lane = threadIdx.x & 31;
    const int colBase = wave * 16;
    const int n  = lane & 15;                  // A: M index / B,C: N index
    const int kg = lane >> 4;                  // K-pair group (0 or 1)

    v8f c = {};
    #pragma unroll
    for (int k = 0; k < IN_FEAT; k += 4) {
        const int kk = k + 2 * kg;
        v2f a, b;
        a.x = ldsA[n][kk];
        a.y = ldsA[n][kk + 1];
        b.x = W[(size_t)kk       * OUTC + colBase + n];
        b.y = W[(size_t)(kk + 1) * OUTC + colBase + n];
        c = __builtin_amdgcn_wmma_f32_16x16x4_f32(
                /*neg_a=*/false, a, /*neg_b=*/false, b,
                /*c_mod=*/(short)0, c, /*reuse_a=*/false, /*reuse_b=*/false);
    }
    #pragma unroll
    for (int r = 0; r < 8; ++r) {
        h[(size_t)(rowBase + r + 8 * kg) * OUTC + colBase + n] = c[r];
    }
}

// ---------------- attention logits per (node, head) ----------------
__global__ void __launch_bounds__(256) gat_att(const float* __restrict__ h,
                                               const float* __restrict__ att_src,
                                               const float* __restrict__ att_dst,
                                               float* __restrict__ ws) {
    int idx = blockIdx.x * 256 + threadIdx.x;
    if (idx >= N_NODES * HEADS) return;
    int node = idx >> 3, hh = idx & 7;
    const float* hp = &h[(size_t)node * OUTC + hh * OUT_CH];
    float s1 = 0.0f, s2 = 0.0f;
    #pragma unroll
    for (int c = 0; c < OUT_CH; ++c) {
        float v = hp[c];
        s1 += v * att_src[hh * OUT_CH + c];
        s2 += v * att_dst[hh * OUT_CH + c];
    }
    ws[WS_ASRC + idx] = s1;
    ws[WS_ADST + idx] = s2;
}

__device__ inline void edge_endpoints(const long long* __restrict__ ei, int e,
                                      int& src, int& dst) {
    if (e < N_EDGES) { src = (int)ei[e]; dst = (int)ei[(size_t)N_EDGES + e]; }
    else             { src = e - N_EDGES; dst = src; }   // self loop
}

// ---------------- scatter max over dst ----------------
__global__ void __launch_bounds__(256) gat_edge_max(const long long* __restrict__ ei,
                                                    float* __restrict__ ws) {
    int idx = blockIdx.x * 256 + threadIdx.x;
    if (idx >= ET * HEADS) return;
    int e = idx >> 3, hh = idx & 7;
    int src, dst; edge_endpoints(ei, e, src, dst);
    float v = ws[WS_ASRC + src * 8 + hh] + ws[WS_ADST + dst * 8 + hh];
    v = v > 0.0f ? v : NEG_SLOPE * v;
    atomicMaxFloat(&ws[WS_MMAX + dst * 8 + hh], v);
}

// ---------------- scatter exp-sum over dst ----------------
__global__ void __launch_bounds__(256) gat_edge_sum(const long long* __restrict__ ei,
                                                    float* __restrict__ ws) {
    int idx = blockIdx.x * 256 + threadIdx.x;
    if (idx >= ET * HEADS) return;
    int e = idx >> 3, hh = idx & 7;
    int src, dst; edge_endpoints(ei, e, src, dst);
    float v = ws[WS_ASRC + src * 8 + hh] + ws[WS_ADST + dst * 8 + hh];
    v = v > 0.0f ? v : NEG_SLOPE * v;
    float ex = expf(v - ws[WS_MMAX + dst * 8 + hh]);
    atomicAdd(&ws[WS_SSUM + dst * 8 + hh], ex);
}

// ---------------- message pass: 16 lanes per edge ----------------
__global__ void __launch_bounds__(256) gat_message(const long long* __restrict__ ei,
                                                   const float* __restrict__ h,
                                                   float* __restrict__ ws) {
    __shared__ float co[16][8];
    const int gid = threadIdx.x >> 4;          // edge slot in block
    const int c   = threadIdx.x & 15;          // output channel
    const int e   = blockIdx.x * 16 + gid;
    const bool valid = (e < ET);
    int src = 0, dst = 0;
    if (valid) {
        edge_endpoints(ei, e, src, dst);
        if (c < 8) {
            int hh = c;
            float v = ws[WS_ASRC + src * 8 + hh] + ws[WS_ADST + dst * 8 + hh];
            v = v > 0.0f ? v : NEG_SLOPE * v;
            float ex = expf(v - ws[WS_MMAX + dst * 8 + hh]);
            co[gid][hh] = ex / ws[WS_SSUM + dst * 8 + hh];
        }
    }
    __syncthreads();
    if (valid) {
        const float* hs = &h[(size_t)src * OUTC + c];
        float val = 0.0f;
        #pragma unroll
        for (int hh = 0; hh < HEADS; ++hh) val += co[gid][hh] * hs[hh * OUT_CH];
        atomicAdd(&ws[WS_ONODE + dst * 16 + c], val * (1.0f / (float)HEADS));
    }
}

// ---------------- bias + ELU + graph pooling accumulate ----------------
__global__ void __launch_bounds__(256) gat_node_final(const long long* __restrict__ batch,
                                                      const float* __restrict__ bias,
                                                      float* __restrict__ ws) {
    int idx = blockIdx.x * 256 + threadIdx.x;
    if (idx >= N_NODES * 16) return;
    int node = idx >> 4, c = idx & 15;
    float v = ws[WS_ONODE + idx] + bias[c];
    v = v > 0.0f ? v : (expf(v) - 1.0f);       // ELU
    int g = (int)batch[node];
    atomicAdd(&ws[WS_GSUM + g * 16 + c], v);
    if (c == 0) atomicAdd(&ws[WS_GCNT + g], 1.0f);
}

__global__ void __launch_bounds__(256) gat_graph_div(const float* __restrict__ ws,
                                                     float* __restrict__ out) {
    int idx = blockIdx.x * 256 + threadIdx.x;
    if (idx >= N_GRAPHS * 16) return;
    out[idx] = ws[WS_GSUM + idx] / fmaxf(ws[WS_GCNT + (idx >> 4)], 1.0f);
}

extern "C" void kernel_launch(void* const* d_in, const int* in_sizes, int n_in,
                              void* d_out, int out_size, void* d_ws, size_t ws_size,
                              hipStream_t stream) {
    (void)in_sizes; (void)n_in; (void)out_size; (void)ws_size;
    const float*     x       = (const float*)d_in[0];
    const long long* ei      = (const long long*)d_in[1];
    const long long* batch   = (const long long*)d_in[2];
    const float*     gamma   = (const float*)d_in[3];
    const float*     beta    = (const float*)d_in[4];
    const float*     W       = (const float*)d_in[5];
    const float*     att_src = (const float*)d_in[6];
    const float*     att_dst = (const float*)d_in[7];
    const float*     bias    = (const float*)d_in[8];
    float* ws  = (float*)d_ws;
    float* out = (float*)d_out;

    float* h = ws + WS_H;

    // 1) init scratch
    gat_init<<<(N_NODES * 16 + 255) / 256, 256, 0, stream>>>(ws);
    // 2) BN statistics (biased variance) and scale/shift fold
    gat_bn_stats<<<(N_NODES + 255) / 256, 128, 0, stream>>>(x, ws);
    gat_bn_finalize<<<1, 128, 0, stream>>>(gamma, beta, ws);
    // 3) fused BN-apply + GEMM (WMMA fp32 16x16x4), h = xn @ W
    gat_gemm_wmma<<<N_NODES / 16, 256, 0, stream>>>(x, W, ws, h);
    // 4) attention logits
    gat_att<<<(N_NODES * HEADS + 255) / 256, 256, 0, stream>>>(h, att_src, att_dst, ws);
    // 5) scatter softmax: max then exp-sum
    gat_edge_max<<<(ET * HEADS + 255) / 256, 256, 0, stream>>>(ei, ws);
    gat_edge_sum<<<(ET * HEADS + 255) / 256, 256, 0, stream>>>(ei, ws);
    // 6) weighted message aggregation (head mean folded in)
    gat_message<<<(ET + 15) / 16, 256, 0, stream>>>(ei, h, ws);
    // 7) bias + ELU + graph pooling
    gat_node_final<<<(N_NODES * 16 + 255) / 256, 256, 0, stream>>>(batch, bias, ws);
    gat_graph_div<<<(N_GRAPHS * 16 + 255) / 256, 256, 0, stream>>>(ws, out);
}